// GNNPolicy_20512763805981
// MI455X (gfx1250) — compile-verified
//
#include <hip/hip_runtime.h>
#include <hip/hip_bf16.h>

typedef __attribute__((ext_vector_type(2))) float v2f;
typedef __attribute__((ext_vector_type(8))) float v8f;

#define NEG_SLOPE 0.2f

__device__ __forceinline__ float lrelu(float x) { return x > 0.0f ? x : NEG_SLOPE * x; }

// monotonic float<->uint encoding for atomicMax-based segment max
__device__ __forceinline__ unsigned fenc(float f) {
    unsigned u = __float_as_uint(f);
    return (u & 0x80000000u) ? ~u : (u | 0x80000000u);
}
__device__ __forceinline__ float fdec(unsigned k) {
    unsigned u = (k & 0x80000000u) ? (k & 0x7fffffffu) : ~k;
    return __uint_as_float(u);
}

// ---------------------------------------------------------------- zero ws
__global__ void k_zero(unsigned* __restrict__ p, size_t n) {
    size_t i = blockIdx.x * (size_t)blockDim.x + threadIdx.x;
    size_t st = (size_t)gridDim.x * blockDim.x;
    for (; i < n; i += st) p[i] = 0u;
}

// ------------------------------------------------- scalars: ea_mean, ce1[4], ce2
__global__ void k_prep(const float* __restrict__ ea, int E,
                       const float* __restrict__ We1, const float* __restrict__ ae1,
                       const float* __restrict__ We2, const float* __restrict__ ae2,
                       float* __restrict__ scal) {
    __shared__ float sbuf[256];
    int t = threadIdx.x;
    float s = 0.0f;
    for (int i = t; i < E; i += 256) s += ea[i];
    sbuf[t] = s;
    __syncthreads();
    for (int w = 128; w > 0; w >>= 1) {
        if (t < w) sbuf[t] += sbuf[t + w];
        __syncthreads();
    }
    if (t == 0) scal[0] = sbuf[0] / (float)E;
    if (t < 4) { float c = 0.0f; for (int i = 0; i < 64; i++) c += We1[t * 64 + i] * ae1[t * 64 + i]; scal[1 + t] = c; }
    if (t == 4) { float c = 0.0f; for (int i = 0; i < 64; i++) c += We2[i] * ae2[i]; scal[5] = c; }
}

// ----------------------------------------- layer1 per-node attention coeffs
__global__ void k_node1(const float* __restrict__ x, const float* __restrict__ W1,
                        const float* __restrict__ as1, const float* __restrict__ ad1,
                        float* __restrict__ a_s1, float* __restrict__ a_d1, int N) {
    int tid = blockIdx.x * blockDim.x + threadIdx.x;
    if (tid >= N * 4) return;
    int n = tid >> 2, h = tid & 3;
    float x0 = x[n * 2], x1 = x[n * 2 + 1];
    float cs = 0.0f, cd = 0.0f;
    for (int c = 0; c < 64; c++) {
        int idx = h * 64 + c;
        float xp = x0 * W1[idx] + x1 * W1[256 + idx];
        cs += xp * as1[idx];
        cd += xp * ad1[idx];
    }
    a_s1[tid] = cs;
    a_d1[tid] = cd;
}

// ------------------------------------------------ layer1 segment max (thread/edge)
__global__ void k_emax1(const int* __restrict__ eidx, const float* __restrict__ ea,
                        const float* __restrict__ a_s1, const float* __restrict__ a_d1,
                        const float* __restrict__ scal, unsigned* __restrict__ max1,
                        int E, int ET) {
    int e = blockIdx.x * blockDim.x + threadIdx.x;
    if (e >= ET) return;
    int s, d; float eav;
    if (e < E) { s = eidx[e]; d = eidx[E + e]; eav = ea[e]; }
    else       { s = d = e - E; eav = scal[0]; }
    for (int h = 0; h < 4; h++) {
        float lg = lrelu(a_s1[s * 4 + h] + a_d1[d * 4 + h] + eav * scal[1 + h]);
        atomicMax(&max1[d * 4 + h], fenc(lg));
    }
}

// -------------------------------- layer1 exp + message accumulate (wave/edge)
__global__ void k_eacc1(const int* __restrict__ eidx, const float* __restrict__ ea,
                        const float* __restrict__ x, const float* __restrict__ W1,
                        const float* __restrict__ a_s1, const float* __restrict__ a_d1,
                        const float* __restrict__ scal, const unsigned* __restrict__ max1,
                        float* __restrict__ den1, float* __restrict__ acc1,
                        int E, int ET) {
    int wid = (blockIdx.x * blockDim.x + threadIdx.x) >> 5;
    int lane = threadIdx.x & 31;
    if (wid >= ET) return;
    int s, d; float eav;
    if (wid < E) { s = eidx[wid]; d = eidx[E + wid]; eav = ea[wid]; }
    else         { s = d = wid - E; eav = scal[0]; }
    float ex[4];
    for (int h = 0; h < 4; h++) {
        float lg = lrelu(a_s1[s * 4 + h] + a_d1[d * 4 + h] + eav * scal[1 + h]);
        ex[h] = __expf(lg - fdec(max1[d * 4 + h]));
    }
    if (lane < 4) atomicAdd(&den1[d * 4 + lane], ex[lane]);
    float x0 = x[s * 2], x1 = x[s * 2 + 1];
#pragma unroll
    for (int j = 0; j < 8; j++) {
        int idx = lane + j * 32;          // 0..255 concat channel
        int h = idx >> 6;
        float msg = x0 * W1[idx] + x1 * W1[256 + idx];
        atomicAdd(&acc1[(size_t)d * 256 + idx], ex[h] * msg);
    }
}

// -------------------------------------------- layer1 finalize (in place -> h1)
__global__ void k_fin1(float* __restrict__ acc1, const float* __restrict__ den1,
                       const float* __restrict__ b1, int N) {
    int tid = blockIdx.x * blockDim.x + threadIdx.x;
    if (tid >= N * 256) return;
    int n = tid >> 8, idx = tid & 255, h = idx >> 6;
    float v = acc1[tid] / den1[n * 4 + h] + b1[idx];
    acc1[tid] = v > 0.0f ? v : 0.0f;
}

// ------------------------------------- layer2 GEMM: xp2 = h1 @ W2  (WMMA f32)
// wave per 16-row block, 4 N-tiles (64 cols). B (W2) staged in LDS in packed
// float2(row 2kk, row 2kk+1) form, K staged in two 128-wide chunks (32KB LDS).
__global__ void k_gemm2(const float* __restrict__ h1, const float* __restrict__ W2,
                        float* __restrict__ xp2, int N) {
    __shared__ float2 Bs[4096];              // [kk 0..63][n 0..63], 32KB
    int wave = blockIdx.x * 8 + (threadIdx.x >> 5);
    int lane = threadIdx.x & 31;
    int base = wave * 16;
    bool active = base < N;                  // wave-uniform
    int arow = base + (lane & 15); if (arow >= N) arow = N - 1; if (arow < 0) arow = 0;
    int half = lane >> 4;                    // 0: K=k,k+1  1: K=k+2,k+3
    int ncol = lane & 15;
    const float* ap = h1 + (size_t)arow * 256 + half * 2;
    v8f c0 = {}, c1 = {}, c2 = {}, c3 = {};
    for (int chunk = 0; chunk < 2; chunk++) {
        // cooperative fill: rows [chunk*128, chunk*128+128) of W2, packed by pairs
        for (int i = threadIdx.x; i < 4096; i += 256) {
            int kk = i >> 6, n = i & 63;
            int r = chunk * 128 + 2 * kk;
            Bs[i] = make_float2(W2[r * 64 + n], W2[(r + 1) * 64 + n]);
        }
        __syncthreads();
        if (active) {
            for (int k = 0; k < 128; k += 4) {
                float2 av = *(const float2*)(ap + chunk * 128 + k);
                v2f a; a.x = av.x; a.y = av.y;
                const float2* brow = &Bs[((k >> 1) + half) * 64 + ncol];
                float2 bv; v2f b;
                bv = brow[0];  b.x = bv.x; b.y = bv.y;
                c0 = __builtin_amdgcn_wmma_f32_16x16x4_f32(false, a, false, b, (short)0, c0, false, false);
                bv = brow[16]; b.x = bv.x; b.y = bv.y;
                c1 = __builtin_amdgcn_wmma_f32_16x16x4_f32(false, a, false, b, (short)0, c1, false, false);
                bv = brow[32]; b.x = bv.x; b.y = bv.y;
                c2 = __builtin_amdgcn_wmma_f32_16x16x4_f32(false, a, false, b, (short)0, c2, false, false);
                bv = brow[48]; b.x = bv.x; b.y = bv.y;
                c3 = __builtin_amdgcn_wmma_f32_16x16x4_f32(false, a, false, b, (short)0, c3, false, false);
            }
        }
        __syncthreads();
    }
    if (active) {
#pragma unroll
        for (int j = 0; j < 8; j++) {
            int row = base + j + half * 8;
            if (row < N) {
                size_t o = (size_t)row * 64 + ncol;
                xp2[o]      = c0[j];
                xp2[o + 16] = c1[j];
                xp2[o + 32] = c2[j];
                xp2[o + 48] = c3[j];
            }
        }
    }
}

// -------------------------------------------- layer2 per-node attention coeffs
__global__ void k_node2(const float* __restrict__ xp2, const float* __restrict__ as2,
                        const float* __restrict__ ad2, float* __restrict__ a_s2,
                        float* __restrict__ a_d2, int N) {
    int n = blockIdx.x * blockDim.x + threadIdx.x;
    if (n >= N) return;
    float cs = 0.0f, cd = 0.0f;
    const float* r = xp2 + (size_t)n * 64;
    for (int c = 0; c < 64; c++) { cs += r[c] * as2[c]; cd += r[c] * ad2[c]; }
    a_s2[n] = cs;
    a_d2[n] = cd;
}

// ------------------------------------------------ layer2 segment max
__global__ void k_emax2(const int* __restrict__ eidx, const float* __restrict__ ea,
                        const float* __restrict__ a_s2, const float* __restrict__ a_d2,
                        const float* __restrict__ scal, unsigned* __restrict__ max2,
                        int E, int ET) {
    int e = blockIdx.x * blockDim.x + threadIdx.x;
    if (e >= ET) return;
    int s, d; float eav;
    if (e < E) { s = eidx[e]; d = eidx[E + e]; eav = ea[e]; }
    else       { s = d = e - E; eav = scal[0]; }
    float lg = lrelu(a_s2[s] + a_d2[d] + eav * scal[5]);
    atomicMax(&max2[d], fenc(lg));
}

// --------------------------------- layer2 exp + message accumulate (wave/edge)
__global__ void k_eacc2(const int* __restrict__ eidx, const float* __restrict__ ea,
                        const float* __restrict__ xp2, const float* __restrict__ a_s2,
                        const float* __restrict__ a_d2, const float* __restrict__ scal,
                        const unsigned* __restrict__ max2, float* __restrict__ den2,
                        float* __restrict__ acc2, int E, int ET) {
    int wid = (blockIdx.x * blockDim.x + threadIdx.x) >> 5;
    int lane = threadIdx.x & 31;
    if (wid >= ET) return;
    int s, d; float eav;
    if (wid < E) { s = eidx[wid]; d = eidx[E + wid]; eav = ea[wid]; }
    else         { s = d = wid - E; eav = scal[0]; }
    float lg = lrelu(a_s2[s] + a_d2[d] + eav * scal[5]);
    float ex = __expf(lg - fdec(max2[d]));
    if (lane == 0) atomicAdd(&den2[d], ex);
    const float* r = xp2 + (size_t)s * 64;
    atomicAdd(&acc2[(size_t)d * 64 + lane], ex * r[lane]);
    atomicAdd(&acc2[(size_t)d * 64 + lane + 32], ex * r[lane + 32]);
}

// -------------------------------------------- layer2 finalize (in place -> h)
__global__ void k_fin2(float* __restrict__ acc2, const float* __restrict__ den2,
                       const float* __restrict__ b2, int N) {
    int tid = blockIdx.x * blockDim.x + threadIdx.x;
    if (tid >= N * 64) return;
    int n = tid >> 6, c = tid & 63;
    acc2[tid] = acc2[tid] / den2[n] + b2[c];
}

// -------- edge scorer: scores = relu([h[src];h[dst];ea] @ Ws1 + bs1) @ Ws2 + bs2
// wave per 16 edges; K=128 WMMA GEMM, B (Ws1 rows 0..127) staged packed in LDS;
// ea*Ws1[128,:] folded into fused epilogue (relu + dot Ws2 + lane reduction).
__global__ void k_score(const int* __restrict__ eidx, const float* __restrict__ ea,
                        const float* __restrict__ h, const float* __restrict__ Ws1,
                        const float* __restrict__ bs1, const float* __restrict__ Ws2,
                        const float* __restrict__ bs2, float* __restrict__ out, int E) {
    __shared__ float2 Bs[4096];              // [kk 0..63][n 0..63], 32KB
    // cooperative fill of packed Ws1 rows 0..127 (before any exit: all threads barrier)
    for (int i = threadIdx.x; i < 4096; i += 256) {
        int kk = i >> 6, n = i & 63;
        Bs[i] = make_float2(Ws1[(2 * kk) * 64 + n], Ws1[(2 * kk + 1) * 64 + n]);
    }
    __syncthreads();

    int wave = blockIdx.x * 8 + (threadIdx.x >> 5);
    int lane = threadIdx.x & 31;
    int base = wave * 16;
    if (base >= E) return;                   // wave-uniform
    int er = base + (lane & 15); if (er >= E) er = E - 1;
    int s = eidx[er], d = eidx[E + er];
    const float* rs = h + (size_t)s * 64;
    const float* rd = h + (size_t)d * 64;
    int half = lane >> 4;
    int ncol = lane & 15;
    v8f c0 = {}, c1 = {}, c2 = {}, c3 = {};
    for (int k = 0; k < 128; k += 4) {
        int koff = k + half * 2;
        const float* rp = (koff < 64) ? (rs + koff) : (rd + (koff - 64));
        float2 av = *(const float2*)rp;
        v2f a; a.x = av.x; a.y = av.y;
        const float2* brow = &Bs[((k >> 1) + half) * 64 + ncol];
        float2 bv; v2f b;
        bv = brow[0];  b.x = bv.x; b.y = bv.y;
        c0 = __builtin_amdgcn_wmma_f32_16x16x4_f32(false, a, false, b, (short)0, c0, false, false);
        bv = brow[16]; b.x = bv.x; b.y = bv.y;
        c1 = __builtin_amdgcn_wmma_f32_16x16x4_f32(false, a, false, b, (short)0, c1, false, false);
        bv = brow[32]; b.x = bv.x; b.y = bv.y;
        c2 = __builtin_amdgcn_wmma_f32_16x16x4_f32(false, a, false, b, (short)0, c2, false, false);
        bv = brow[48]; b.x = bv.x; b.y = bv.y;
        c3 = __builtin_amdgcn_wmma_f32_16x16x4_f32(false, a, false, b, (short)0, c3, false, false);
    }
    // epilogue: bias + ea*Ws1[128,:], relu, dot Ws2, reduce over 16 lanes/half
    float w20 = Ws2[ncol], w21 = Ws2[16 + ncol], w22 = Ws2[32 + ncol], w23 = Ws2[48 + ncol];
    float bb0 = bs1[ncol], bb1 = bs1[16 + ncol], bb2 = bs1[32 + ncol], bb3 = bs1[48 + ncol];
    float we0 = Ws1[128 * 64 + ncol], we1 = Ws1[128 * 64 + 16 + ncol];
    float we2 = Ws1[128 * 64 + 32 + ncol], we3 = Ws1[128 * 64 + 48 + ncol];
#pragma unroll
    for (int j = 0; j < 8; j++) {
        int erj = base + j + half * 8;       // this lane's edge for vgpr j
        float eav = (erj < E) ? ea[erj] : 0.0f;
        float v, sum = 0.0f;
        v = c0[j] + bb0 + eav * we0; sum += (v > 0.0f ? v : 0.0f) * w20;
        v = c1[j] + bb1 + eav * we1; sum += (v > 0.0f ? v : 0.0f) * w21;
        v = c2[j] + bb2 + eav * we2; sum += (v > 0.0f ? v : 0.0f) * w22;
        v = c3[j] + bb3 + eav * we3; sum += (v > 0.0f ? v : 0.0f) * w23;
        sum += __shfl_xor(sum, 1, 32);
        sum += __shfl_xor(sum, 2, 32);
        sum += __shfl_xor(sum, 4, 32);
        sum += __shfl_xor(sum, 8, 32);
        if (ncol == 0 && erj < E) out[erj] = sum + bs2[0];
    }
}

extern "C" void kernel_launch(void* const* d_in, const int* in_sizes, int n_in,
                              void* d_out, int out_size, void* d_ws, size_t ws_size,
                              hipStream_t stream) {
    const float* x   = (const float*)d_in[0];
    const int*   ei  = (const int*)d_in[1];
    const float* ea  = (const float*)d_in[2];
    const float* W1  = (const float*)d_in[3];
    const float* as1 = (const float*)d_in[4];
    const float* ad1 = (const float*)d_in[5];
    const float* We1 = (const float*)d_in[6];
    const float* ae1 = (const float*)d_in[7];
    const float* b1  = (const float*)d_in[8];
    const float* W2  = (const float*)d_in[9];
    const float* as2 = (const float*)d_in[10];
    const float* ad2 = (const float*)d_in[11];
    const float* We2 = (const float*)d_in[12];
    const float* ae2 = (const float*)d_in[13];
    const float* b2  = (const float*)d_in[14];
    const float* Ws1 = (const float*)d_in[15];
    const float* bs1 = (const float*)d_in[16];
    const float* Ws2 = (const float*)d_in[17];
    const float* bs2 = (const float*)d_in[18];
    float* out = (float*)d_out;

    const int N  = in_sizes[0] / 2;
    const int E  = in_sizes[2];
    const int ET = N + E;

    // ---- workspace carve (256B aligned regions) ----
    char* ws = (char*)d_ws;
    size_t off = 0;
    auto carve = [&](size_t nbytes) -> char* {
        char* p = ws + off;
        off = (off + nbytes + 255) & ~(size_t)255;
        return p;
    };
    float*    scal = (float*)carve(64);                      // ea_mean, ce1[4], ce2
    float*    a_s1 = (float*)carve((size_t)N * 4 * 4);
    float*    a_d1 = (float*)carve((size_t)N * 4 * 4);
    unsigned* max1 = (unsigned*)carve((size_t)N * 4 * 4);
    float*    den1 = (float*)carve((size_t)N * 4 * 4);
    float*    acc1 = (float*)carve((size_t)N * 256 * 4);     // -> h1 in place
    float*    xp2  = (float*)carve((size_t)N * 64 * 4);
    float*    a_s2 = (float*)carve((size_t)N * 4);
    float*    a_d2 = (float*)carve((size_t)N * 4);
    unsigned* max2 = (unsigned*)carve((size_t)N * 4);
    float*    den2 = (float*)carve((size_t)N * 4);
    float*    acc2 = (float*)carve((size_t)N * 64 * 4);      // -> h in place
    size_t total_dw = off / 4;

    const int TB = 256;

    // 0) zero whole used workspace (max keys: encoded -inf > 0, so 0 is safe init)
    k_zero<<<2048, TB, 0, stream>>>((unsigned*)ws, total_dw);
    // 1) scalar precompute
    k_prep<<<1, TB, 0, stream>>>(ea, E, We1, ae1, We2, ae2, scal);
    // 2) layer1 node coeffs
    k_node1<<<(N * 4 + TB - 1) / TB, TB, 0, stream>>>(x, W1, as1, ad1, a_s1, a_d1, N);
    // 3) layer1 segment max
    k_emax1<<<(ET + TB - 1) / TB, TB, 0, stream>>>(ei, ea, a_s1, a_d1, scal, max1, E, ET);
    // 4) layer1 exp + accumulate (wave per edge)
    k_eacc1<<<(ET + 7) / 8, TB, 0, stream>>>(ei, ea, x, W1, a_s1, a_d1, scal, max1, den1, acc1, E, ET);
    // 5) layer1 finalize -> h1 (in acc1)
    k_fin1<<<(N * 256 + TB - 1) / TB, TB, 0, stream>>>(acc1, den1, b1, N);
    // 6) layer2 GEMM xp2 = h1 @ W2 (WMMA, LDS-staged B)
    {
        int waves = (N + 15) / 16;
        k_gemm2<<<(waves + 7) / 8, TB, 0, stream>>>(acc1, W2, xp2, N);
    }
    // 7) layer2 node coeffs
    k_node2<<<(N + TB - 1) / TB, TB, 0, stream>>>(xp2, as2, ad2, a_s2, a_d2, N);
    // 8) layer2 segment max
    k_emax2<<<(ET + TB - 1) / TB, TB, 0, stream>>>(ei, ea, a_s2, a_d2, scal, max2, E, ET);
    // 9) layer2 exp + accumulate
    k_eacc2<<<(ET + 7) / 8, TB, 0, stream>>>(ei, ea, xp2, a_s2, a_d2, scal, max2, den2, acc2, E, ET);
    // 10) layer2 finalize -> h (in acc2)
    k_fin2<<<(N * 64 + TB - 1) / TB, TB, 0, stream>>>(acc2, den2, b2, N);
    // 11) edge scorer (WMMA, LDS-staged B, fused epilogue)
    {
        int waves = (E + 15) / 16;
        k_score<<<(waves + 7) / 8, TB, 0, stream>>>(ei, ea, acc2, Ws1, bs1, Ws2, bs2, out, E);
    }
}